// SURVEY_shapes_Cheb_81638738363109
// MI455X (gfx1250) — compile-verified
//
#include <hip/hip_runtime.h>
#include <hip/hip_bf16.h>

typedef float v2f __attribute__((ext_vector_type(2)));
typedef float v8f __attribute__((ext_vector_type(8)));

#define LDP 32  // padded feature width (HID=30, F_IN=10 both fit)

// ---------------- utility ----------------
__global__ void k_zero(float* __restrict__ p, size_t n) {
  size_t i = (size_t)blockIdx.x * blockDim.x + threadIdx.x;
  if (i < n) p[i] = 0.f;
}

// ---------------- cheb norm ----------------
__global__ void k_deg(const int* __restrict__ ei, const float* __restrict__ ew,
                      float* __restrict__ deg, int nE) {
  int e = blockIdx.x * blockDim.x + threadIdx.x;
  if (e >= nE) return;
  __builtin_prefetch(&ei[e + 4096], 0, 1);
  int s = ei[e], d = ei[nE + e];
  float w = (s == d) ? 0.f : ew[e];
  atomicAdd(&deg[s], w);
}

__global__ void k_dinv(float* __restrict__ deg, int n) {
  int i = blockIdx.x * blockDim.x + threadIdx.x;
  if (i >= n) return;
  float d = deg[i];
  deg[i] = (d > 0.f) ? rsqrtf(fmaxf(d, 1e-30f)) : 0.f;
}

__global__ void k_norm(const int* __restrict__ ei, const float* __restrict__ ew,
                       const float* __restrict__ dinv, float* __restrict__ norm, int nE) {
  int e = blockIdx.x * blockDim.x + threadIdx.x;
  if (e >= nE) return;
  int s = ei[e], d = ei[nE + e];
  float w = (s == d) ? 0.f : ew[e];
  norm[e] = -dinv[s] * w * dinv[d];
}

// ---------------- feature buffers ----------------
__global__ void k_pad_copy(const float* __restrict__ x, float* __restrict__ o,
                           size_t nTot, int F) {
  size_t idx = (size_t)blockIdx.x * blockDim.x + threadIdx.x;
  if (idx >= nTot) return;
  int j = (int)(idx & (LDP - 1));
  size_t i = idx >> 5;
  o[idx] = (j < F) ? x[i * F + j] : 0.f;
}

// out[dst] += norm * in[src]   (padded LDP rows; only first F cols carry data)
__global__ void k_prop(const float* __restrict__ in, float* __restrict__ out,
                       const int* __restrict__ ei, const float* __restrict__ norm,
                       int nE, int F) {
  int e = blockIdx.x * blockDim.x + threadIdx.x;
  if (e >= nE) return;
  __builtin_prefetch(&ei[e + 4096], 0, 1);
  __builtin_prefetch(&norm[e + 4096], 0, 1);
  int s = ei[e], d = ei[nE + e];
  float nw = norm[e];
  const float* hs = in + (size_t)s * LDP;
  float* od = out + (size_t)d * LDP;
  for (int j = 0; j < F; ++j) atomicAdd(&od[j], nw * hs[j]);
}

// D = 2*D - T0 (elementwise over padded buffer)
__global__ void k_combine(float* __restrict__ D, const float* __restrict__ T0, size_t n) {
  size_t i = (size_t)blockIdx.x * blockDim.x + threadIdx.x;
  if (i < n) D[i] = 2.f * D[i] - T0[i];
}

// h = relu(acc + b) with zeroed padding cols
__global__ void k_bias_relu(const float* __restrict__ acc, const float* __restrict__ b,
                            float* __restrict__ h, size_t nTot, int nOut) {
  size_t idx = (size_t)blockIdx.x * blockDim.x + threadIdx.x;
  if (idx >= nTot) return;
  int j = (int)(idx & (LDP - 1));
  float v = 0.f;
  if (j < nOut) v = fmaxf(acc[idx] + b[j], 0.f);
  h[idx] = v;
}

// ---------------- WMMA f32 GEMM: acc(16-row tile x 32 cols) += T @ W ----------------
// FP = padded K extent (multiple of 4, <= LDP), fully unrolled.
// W is staged into LDS as K-pair-interleaved float2 (zero-padded): one aligned
// ds_load_b64 per B fragment, data lands directly in an even VGPR pair.
template <int FP>
__global__ void k_gemm_wmma(const float* __restrict__ T, const float* __restrict__ W,
                            float* __restrict__ acc, int nNodes, int F, int nOut) {
#if defined(__HIP_DEVICE_COMPILE__)
#if !__has_builtin(__builtin_amdgcn_wmma_f32_16x16x4_f32)
#error "no __builtin_amdgcn_wmma_f32_16x16x4_f32 on this device toolchain"
#endif
  __shared__ v2f Wp[(LDP / 2) * LDP];  // 4 KB: Wp[kp*32+c] = {W[2kp][c], W[2kp+1][c]}
  for (int idx = (int)threadIdx.x; idx < (LDP / 2) * LDP; idx += (int)blockDim.x) {
    int kp = idx >> 5, c = idx & (LDP - 1);
    int k0 = kp * 2, k1 = k0 + 1;
    v2f w;
    w.x = (k0 < F && c < nOut) ? W[k0 * nOut + c] : 0.f;
    w.y = (k1 < F && c < nOut) ? W[k1 * nOut + c] : 0.f;
    Wp[idx] = w;
  }
  __syncthreads();

  // one wave32 per 16-row node tile; two 16x16 C fragments cover 32 padded cols
  int wave = blockIdx.x * (blockDim.x >> 5) + ((int)threadIdx.x >> 5);
  int lane = (int)threadIdx.x & 31;
  int row0 = wave << 4;
  if (row0 >= nNodes) return;          // wave-uniform: EXEC stays all-ones for WMMA

  int hi  = lane >> 4;                 // half-wave select
  int l15 = lane & 15;
  const float* Trow = T + (size_t)(row0 + l15) * LDP;  // A: lane = row M
  int koff = hi * 2;                   // A VGPR0/1 hold K=koff, koff+1
  int crow = row0 + hi * 8;            // C: VGPR v -> M = v + 8*hi

  v8f c0, c1;
#pragma unroll
  for (int v = 0; v < 8; ++v) {
    size_t base = (size_t)(crow + v) * LDP + l15;
    c0[v] = acc[base];
    c1[v] = acc[base + 16];
  }

#pragma unroll
  for (int k0 = 0; k0 < FP; k0 += 4) {
    int kr0 = k0 + koff;                    // even
    int kp  = kr0 >> 1;
    v2f a  = *(const v2f*)(Trow + kr0);     // 8B-aligned; padded cols are zero
    v2f b0 = Wp[kp * LDP + l15];            // {W[kr0][c], W[kr0+1][c]}
    v2f b1 = Wp[kp * LDP + 16 + l15];
    c0 = __builtin_amdgcn_wmma_f32_16x16x4_f32(false, a, false, b0, (short)0, c0, false, false);
    c1 = __builtin_amdgcn_wmma_f32_16x16x4_f32(false, a, false, b1, (short)0, c1, false, false);
  }

#pragma unroll
  for (int v = 0; v < 8; ++v) {
    size_t base = (size_t)(crow + v) * LDP + l15;
    acc[base] = c0[v];
    acc[base + 16] = c1[v];
  }
#endif  // __HIP_DEVICE_COMPILE__
}

// ---------------- classifier head ----------------
__global__ void k_final(const float* __restrict__ h, const float* __restrict__ Wl,
                        const float* __restrict__ bl, float* __restrict__ out,
                        int nNodes, int nOut, int nCls) {
  int i = blockIdx.x * blockDim.x + threadIdx.x;
  if (i >= nNodes) return;
  float s[8];
  for (int c = 0; c < nCls; ++c) s[c] = bl[c];
  const float* hr = h + (size_t)i * LDP;
  for (int j = 0; j < nOut; ++j) {
    float hv = hr[j];
    for (int c = 0; c < nCls; ++c) s[c] += hv * Wl[j * nCls + c];
  }
  for (int c = 0; c < nCls; ++c) out[(size_t)i * nCls + c] = s[c];
}

// ---------------- host orchestration ----------------
static inline void launch_gemm(const float* T, const float* W, float* acc,
                               int nNodes, int F, int nOut,
                               unsigned gW, int B, hipStream_t stream) {
  int Fp = (F + 3) & ~3;
  if (Fp <= 12)
    k_gemm_wmma<12><<<gW, B, 0, stream>>>(T, W, acc, nNodes, F, nOut);
  else if (Fp <= 16)
    k_gemm_wmma<16><<<gW, B, 0, stream>>>(T, W, acc, nNodes, F, nOut);
  else
    k_gemm_wmma<32><<<gW, B, 0, stream>>>(T, W, acc, nNodes, F, nOut);
}

extern "C" void kernel_launch(void* const* d_in, const int* in_sizes, int n_in,
                              void* d_out, int out_size, void* d_ws, size_t ws_size,
                              hipStream_t stream) {
  const float* x  = (const float*)d_in[0];
  const int*   ei = (const int*)d_in[1];
  const float* ew = (const float*)d_in[2];
  const float* W1 = (const float*)d_in[3];
  const float* b1 = (const float*)d_in[4];
  const float* W2 = (const float*)d_in[5];
  const float* b2 = (const float*)d_in[6];
  const float* Wl = (const float*)d_in[7];
  const float* bl = (const float*)d_in[8];
  float* out = (float*)d_out;

  const int HID    = in_sizes[4];
  const int NCLS   = in_sizes[8];
  const int nNodes = out_size / NCLS;
  const int FIN    = in_sizes[0] / nNodes;
  const int nE     = in_sizes[2];
  const int K1     = in_sizes[3] / (FIN * HID);
  const int K2     = in_sizes[5] / (HID * HID);

  float* ws   = (float*)d_ws;
  float* norm = ws;                    // nE
  float* deg  = norm + nE;             // nNodes (reused as dinv)
  const size_t P = (size_t)nNodes * LDP;
  float* b0p = deg + nNodes;
  float* b1p = b0p + P;
  float* b2p = b1p + P;
  float* acc = b2p + P;

  const int B = 256;
  auto g = [](size_t n, int b) { return (unsigned)((n + (size_t)b - 1) / b); };
  const unsigned gW = g((size_t)((nNodes + 15) / 16), B / 32);  // 8 waves/block

  // --- normalization: deg -> d^{-1/2} -> edge norm ---
  k_zero<<<g(nNodes, B), B, 0, stream>>>(deg, (size_t)nNodes);
  k_deg <<<g(nE, B),     B, 0, stream>>>(ei, ew, deg, nE);
  k_dinv<<<g(nNodes, B), B, 0, stream>>>(deg, nNodes);
  k_norm<<<g(nE, B),     B, 0, stream>>>(ei, ew, deg, norm, nE);

  // --- layer 1 (F=FIN) ---
  float* t0 = b0p; float* t1 = b1p; float* t2 = b2p;
  k_pad_copy<<<g(P, B), B, 0, stream>>>(x, t0, P, FIN);
  k_zero<<<g(P, B), B, 0, stream>>>(acc, P);
  launch_gemm(t0, W1, acc, nNodes, FIN, HID, gW, B, stream);
  k_zero<<<g(P, B), B, 0, stream>>>(t1, P);
  k_prop<<<g(nE, B), B, 0, stream>>>(t0, t1, ei, norm, nE, FIN);
  launch_gemm(t1, W1 + (size_t)FIN * HID, acc, nNodes, FIN, HID, gW, B, stream);
  for (int k = 2; k < K1; ++k) {
    k_zero<<<g(P, B), B, 0, stream>>>(t2, P);
    k_prop<<<g(nE, B), B, 0, stream>>>(t1, t2, ei, norm, nE, FIN);
    k_combine<<<g(P, B), B, 0, stream>>>(t2, t0, P);
    launch_gemm(t2, W1 + (size_t)k * FIN * HID, acc, nNodes, FIN, HID, gW, B, stream);
    float* tmp = t0; t0 = t1; t1 = t2; t2 = tmp;
  }
  float* h1 = t2;  // free buffer after rotation
  k_bias_relu<<<g(P, B), B, 0, stream>>>(acc, b1, h1, P, HID);

  // --- layer 2 (F=HID) ---
  float* u0 = h1; float* u1 = t0; float* u2 = t1;
  k_zero<<<g(P, B), B, 0, stream>>>(acc, P);
  launch_gemm(u0, W2, acc, nNodes, HID, HID, gW, B, stream);
  k_zero<<<g(P, B), B, 0, stream>>>(u1, P);
  k_prop<<<g(nE, B), B, 0, stream>>>(u0, u1, ei, norm, nE, HID);
  launch_gemm(u1, W2 + (size_t)HID * HID, acc, nNodes, HID, HID, gW, B, stream);
  for (int k = 2; k < K2; ++k) {
    k_zero<<<g(P, B), B, 0, stream>>>(u2, P);
    k_prop<<<g(nE, B), B, 0, stream>>>(u1, u2, ei, norm, nE, HID);
    k_combine<<<g(P, B), B, 0, stream>>>(u2, u0, P);
    launch_gemm(u2, W2 + (size_t)k * HID * HID, acc, nNodes, HID, HID, gW, B, stream);
    float* tmp = u0; u0 = u1; u1 = u2; u2 = tmp;
  }
  float* h2 = u2;
  k_bias_relu<<<g(P, B), B, 0, stream>>>(acc, b2, h2, P, HID);

  // --- classifier head ---
  k_final<<<g(nNodes, B), B, 0, stream>>>(h2, Wl, bl, out, nNodes, HID, NCLS);
}